// H2GCNConv_24438363914374
// MI455X (gfx1250) — compile-verified
//
#include <hip/hip_runtime.h>

// H2GCN dual-SpMM scatter kernel for MI455X (gfx1250, wave32).  FINAL FORM.
//
// out[dst1, 0:64]   += w1 * x[src1]   (E1 = 1.6M edges)
// out[dst2, 64:128] += w2 * x[src2]   (E2 = 3.2M edges)
//
// Memory/atomic bound: 0.61 GFLOP vs ~2.5 GB of L2 traffic (AI ~ 0.24 f/B,
// ~4 orders of magnitude below the WMMA crossover -> WMMA inapplicable);
// the ~173 MB working set fits the 192 MB L2. gfx1250 mapping:
//   - edge indices: wave-uniform 128-bit scalar-path fetches (SMEM/KMcnt)
//   - source-row gather: TDM gather mode (tensor_load_to_lds, TENSORcnt),
//     two 8-row x 256B descriptors per wave, double-buffered via the split
//     wait (s_wait_tensorcnt 1 -> scatter half A while half B streams)
//   - scatter: no-return global_atomic_add_f32 resolved at the L2 atomic units
//   - wave32 x float2 = one full 64-wide row per wave op

typedef unsigned int       u32;
typedef unsigned long long u64;
typedef __attribute__((ext_vector_type(4))) u32 v4u;
typedef __attribute__((ext_vector_type(8))) int v8i;
typedef __attribute__((ext_vector_type(4))) int v4i;

constexpr int D          = 64;    // feature dim (row = 256 B)
constexpr int OUT_STRIDE = 128;   // concat([x1, x2], axis=1)
constexpr int HALF       = 8;     // rows per TDM gather descriptor (32-bit idx max)
constexpr int EPW        = 16;    // edges per wave = 2 descriptors
constexpr int WPB        = 8;     // waves per 256-thread block

__global__ __launch_bounds__(256) void zero_f4(float4* __restrict__ out, int n4) {
  int i = blockIdx.x * blockDim.x + threadIdx.x;
  if (i < n4) out[i] = make_float4(0.f, 0.f, 0.f, 0.f);
}

// Issue one gather-mode TDM op from a prebuilt descriptor template, patching
// only lds_addr (g0[1]) and tile_dim1 = #valid indices (g1[4]).
__device__ __forceinline__ void tdm_gather_rows(v4u g0, v8i g1, u32 lds_off,
                                                int nvalid, v4i idx_lo, v4i idx_hi) {
  g0.y  = lds_off;           // D# group0: lds_addr (bytes)
  g1[4] = nvalid & 0xFFFF;   // D# group1: tile_dim1 = #valid gather indices
#if defined(__clang_major__) && __clang_major__ >= 23
  const v8i gz = { 0, 0, 0, 0, 0, 0, 0, 0 };
  __builtin_amdgcn_tensor_load_to_lds(g0, g1, idx_lo, idx_hi, gz, 0);
#else
  __builtin_amdgcn_tensor_load_to_lds(g0, g1, idx_lo, idx_hi, 0);
#endif
}

__global__ __launch_bounds__(256) void spmm_scatter_tdm(
    const float* __restrict__ x,
    const int*   __restrict__ src,
    const int*   __restrict__ dst,
    const float* __restrict__ w,
    float*       __restrict__ out,
    int E, int n_nodes, int col_off) {
  __shared__ float smem[WPB * EPW * D];          // 32 KB: 4 KB slice per wave

  const int lane = threadIdx.x & 31;             // wave32
  const int wib  = __builtin_amdgcn_readfirstlane((int)(threadIdx.x >> 5));
  const int e0   = (blockIdx.x * WPB + wib) * EPW;
  if (e0 >= E) return;

  int   se[EPW], de[EPW];
  float we[EPW];
  if (e0 + EPW <= E) {
    // e0 % 16 == 0 -> aligned 128-bit uniform fetches (scalar path).
#pragma unroll
    for (int q = 0; q < EPW / 4; ++q) {
      const int4   s4 = *(const int4*)  (src + e0 + 4 * q);
      const int4   d4 = *(const int4*)  (dst + e0 + 4 * q);
      const float4 w4 = *(const float4*)(w   + e0 + 4 * q);
      se[4*q+0]=s4.x; se[4*q+1]=s4.y; se[4*q+2]=s4.z; se[4*q+3]=s4.w;
      de[4*q+0]=d4.x; de[4*q+1]=d4.y; de[4*q+2]=d4.z; de[4*q+3]=d4.w;
      we[4*q+0]=w4.x; we[4*q+1]=w4.y; we[4*q+2]=w4.z; we[4*q+3]=w4.w;
    }
  } else {
#pragma unroll
    for (int k = 0; k < EPW; ++k) {
      const int e  = e0 + k;
      const bool ok = (e < E);
      se[k] = ok ? src[e] : 0;
      de[k] = ok ? dst[e] : -1;   // -1 marks inactive tail edge
      we[k] = ok ? w[e]   : 0.f;
    }
  }

  // Guarantee SGPR residency for the descriptor payload.
#pragma unroll
  for (int k = 0; k < EPW; ++k) se[k] = __builtin_amdgcn_readfirstlane(se[k]);

  const int rem = E - e0;
  const int nv1 = rem < HALF ? rem : HALF;                 // >= 1
  const int r2  = rem - HALF;
  const int nv2 = r2 < 0 ? 0 : (r2 > HALF ? HALF : r2);

  float* lds0 = smem + (size_t)wib * (EPW * D);
  float* lds1 = lds0 + HALF * D;

  // ---- Descriptor template, built once (CDNA5 ISA) ----
  const u64 ga = (u64)(size_t)x;                 // tensor base (rows are indexed)
  v4u g0;
  g0.x = 1u            // count = 1 (valid descriptor)
       | (1u << 30)    // gather_index_size = 1 -> 32-bit row indices
       | (1u << 31);   // gather_mode = 1
  g0.y = 0;                                             // lds_addr patched per op
  g0.z = (u32)ga;                                       // global_addr[31:0]
  g0.w = ((u32)(ga >> 32) & 0x01FFFFFFu) | 0x80000000u; // addr[56:32] | type=2
  v8i g1;
  g1[0] = 0x20000;                                  // data_size=2 (4 B); no mcast
  g1[1] = (D & 0xFFFF) << 16;                       // tensor_dim0[15:0] @ [63:48]
  g1[2] = (int)(((u32)n_nodes & 0xFFFFu) << 16);    // tensor_dim1[15:0] @ [95:80]
  g1[3] = (D << 16) | ((n_nodes >> 16) & 0xFFFF);   // tile_dim0 | tensor_dim1 hi
  g1[4] = 0;                                        // tile_dim1 patched per op
  g1[5] = D;                                        // tensor_dim0_stride lo32
  g1[6] = 0;
  g1[7] = 0;

  // Issue both gathers back-to-back; TDM ops of a wave complete in order.
  tdm_gather_rows(g0, g1, (u32)(size_t)lds0, nv1,
                  v4i{se[0], se[1], se[2], se[3]},
                  v4i{se[4], se[5], se[6], se[7]});
  if (nv2 > 0)
    tdm_gather_rows(g0, g1, (u32)(size_t)lds1, nv2,
                    v4i{se[8],  se[9],  se[10], se[11]},
                    v4i{se[12], se[13], se[14], se[15]});

  // Scatter 8 rows from an LDS slice: 32 lanes x float2 = full row per op.
  auto scatter8 = [&](const float* lds, int base) {
#pragma unroll
    for (int k = 0; k < HALF; ++k) {
      const int kk = base + k;
      if (de[kk] >= 0) {
        const float2 v = *(const float2*)(lds + k * D + lane * 2);
        float* o = out + (size_t)de[kk] * OUT_STRIDE + col_off + (size_t)lane * 2;
        unsafeAtomicAdd(o + 0, we[kk] * v.x);   // no-return global_atomic_add_f32
        unsafeAtomicAdd(o + 1, we[kk] * v.y);
      }
    }
  };

  if (nv2 > 0) {
    // Half A has landed once <=1 TDM op remains outstanding (in-order per
    // wave); scatter it while half B is still streaming into LDS.
    __builtin_amdgcn_s_wait_tensorcnt(1);
    scatter8(lds0, 0);
    __builtin_amdgcn_s_wait_tensorcnt(0);
    scatter8(lds1, HALF);
  } else {
    __builtin_amdgcn_s_wait_tensorcnt(0);
    scatter8(lds0, 0);
  }
}

extern "C" void kernel_launch(void* const* d_in, const int* in_sizes, int n_in,
                              void* d_out, int out_size, void* d_ws, size_t ws_size,
                              hipStream_t stream) {
  const float* x    = (const float*)d_in[0];
  const int*   src1 = (const int*)  d_in[1];
  const int*   dst1 = (const int*)  d_in[2];
  const float* w1   = (const float*)d_in[3];
  const int*   src2 = (const int*)  d_in[4];
  const int*   dst2 = (const int*)  d_in[5];
  const float* w2   = (const float*)d_in[6];
  float* out = (float*)d_out;

  const int E1 = in_sizes[1];
  const int E2 = in_sizes[4];
  const int n_nodes = in_sizes[0] / D;

  // Atomic accumulation requires a clean output every call.
  const int n4 = out_size / 4;
  zero_f4<<<(n4 + 255) / 256, 256, 0, stream>>>((float4*)out, n4);

  auto blocks_for = [](int E) {
    const long long nwaves = ((long long)E + EPW - 1) / EPW;
    return (int)((nwaves + WPB - 1) / WPB);
  };

  spmm_scatter_tdm<<<blocks_for(E1), 256, 0, stream>>>(x, src1, dst1, w1, out,
                                                       E1, n_nodes, 0);
  spmm_scatter_tdm<<<blocks_for(E2), 256, 0, stream>>>(x, src2, dst2, w2, out,
                                                       E2, n_nodes, D);
}